// GcnLayers_2559800508644
// MI455X (gfx1250) — compile-verified
//
#include <hip/hip_runtime.h>
#include <hip/hip_bf16.h>
#include <math.h>

// CDNA5 / gfx1250 wave32 WMMA types
typedef __attribute__((ext_vector_type(2))) float v2f;
typedef __attribute__((ext_vector_type(8))) float v8f;

#define D_FEAT 128
#define EPS_BN 1e-5f

// ---------------------------------------------------------------------------
// GEMM: H[N,128] = X[N,128] @ W^T[128,128] + bias
// One wave per 16x16 output tile, V_WMMA_F32_16X16X4_F32, K=128 -> 32 WMMAs.
// block = 256 threads = 8 waves -> 8 column tiles (covers all 128 cols),
// grid.x = N/16 row tiles. Requires N % 16 == 0 for this kernel (remainder
// rows handled by gemm_rem_kernel).
// ---------------------------------------------------------------------------
__global__ void gemm_bias_wmma_kernel(const float* __restrict__ X,
                                      const float* __restrict__ W,
                                      const float* __restrict__ bias,
                                      float* __restrict__ H,
                                      int nTiles) {
    const int lane = threadIdx.x & 31;
    const int wave = threadIdx.x >> 5;            // 0..7 : column tile
    const int rowTile = blockIdx.x;
    if (rowTile >= nTiles) return;

    const int rowBase = rowTile * 16;
    const int colBase = wave * 16;
    const int half = lane >> 4;                   // 0: lanes 0-15, 1: lanes 16-31
    const int l15  = lane & 15;

    // A: row = rowBase + l15, this lane holds K pair {k0+2*half, k0+2*half+1}
    const float* aPtr = X + (size_t)(rowBase + l15) * D_FEAT + half * 2;
    // B[k][n] = W[n][k] (x @ W^T): col n = colBase + l15, same K pair striping
    const float* bPtr = W + (size_t)(colBase + l15) * D_FEAT + half * 2;

    v8f acc = {};
#pragma unroll
    for (int k = 0; k < D_FEAT; k += 4) {
        v2f a = *(const v2f*)(aPtr + k);
        v2f b = *(const v2f*)(bPtr + k);
        // D = A*B + C ; 8 args: (neg_a, A, neg_b, B, c_mod, C, reuse_a, reuse_b)
        acc = __builtin_amdgcn_wmma_f32_16x16x4_f32(
            false, a, false, b, (short)0, acc, false, false);
    }

    const float bv = bias[colBase + l15];
    // C/D layout: VGPR v -> row (v + 8*half), lane l15 -> column
#pragma unroll
    for (int v = 0; v < 8; ++v) {
        const int r = rowBase + v + 8 * half;
        H[(size_t)r * D_FEAT + colBase + l15] = acc[v] + bv;
    }
}

// Scalar fallback for remainder rows (N % 16). Not used for N=50000 but keeps
// the launch generic. One thread per (row, col).
__global__ void gemm_rem_kernel(const float* __restrict__ X,
                                const float* __restrict__ W,
                                const float* __restrict__ bias,
                                float* __restrict__ H,
                                int rowStart, int nRows) {
    int t = blockIdx.x * blockDim.x + threadIdx.x;
    int total = (nRows - rowStart) * D_FEAT;
    if (t >= total) return;
    int r = rowStart + t / D_FEAT;
    int c = t % D_FEAT;
    const float* x = X + (size_t)r * D_FEAT;
    const float* w = W + (size_t)c * D_FEAT;
    float s = bias[c];
    for (int k = 0; k < D_FEAT; ++k) s = fmaf(x[k], w[k], s);
    H[(size_t)r * D_FEAT + c] = s;
}

// ---------------------------------------------------------------------------
// SpMM scatter: AGG[dst] += val * H[src]. One wave32 per edge; each lane owns
// a float4 (4 features). agg fits in the 192MB L2, so global fp32 atomics
// resolve on-die.
// ---------------------------------------------------------------------------
__global__ void scatter_kernel(const float* __restrict__ H,
                               const int* __restrict__ esrc,
                               const int* __restrict__ edst,
                               const float* __restrict__ eval,
                               float* __restrict__ AGG,
                               int E) {
    const int lane = threadIdx.x & 31;
    const int e = blockIdx.x * 8 + (threadIdx.x >> 5);
    if (e >= E) return;
    const int s = esrc[e];
    const int d = edst[e];
    const float v = eval[e];
    const float4 hv = *((const float4*)(H + (size_t)s * D_FEAT) + lane);
    float* op = AGG + (size_t)d * D_FEAT + lane * 4;
    atomicAdd(op + 0, v * hv.x);
    atomicAdd(op + 1, v * hv.y);
    atomicAdd(op + 2, v * hv.z);
    atomicAdd(op + 3, v * hv.w);
}

// ---------------------------------------------------------------------------
// Fused PReLU (+optional residual) + BN statistics. Block = 128 threads, one
// column each; each block owns a 128-row strip, accumulates in registers,
// then a single atomicAdd per column per block.
// ---------------------------------------------------------------------------
#define ROWS_PER_BLOCK 128
__global__ void prelu_res_stats_kernel(float* __restrict__ Y,          // in: agg, out: activated (in place)
                                       const float* __restrict__ res,  // may be nullptr
                                       const float* __restrict__ alpha_ptr,
                                       float* __restrict__ colSum,
                                       float* __restrict__ colSumSq,
                                       int nRows) {
    const float a = *alpha_ptr;
    const int c = threadIdx.x;                     // 0..127
    int r0 = blockIdx.x * ROWS_PER_BLOCK;
    int r1 = r0 + ROWS_PER_BLOCK;
    if (r1 > nRows) r1 = nRows;
    float s = 0.0f, s2 = 0.0f;
    for (int r = r0; r < r1; ++r) {
        const size_t idx = (size_t)r * D_FEAT + c;
        float v = Y[idx];
        v = v > 0.0f ? v : a * v;                  // PReLU (shared slope)
        if (res) v += res[idx];                    // residual before BN
        Y[idx] = v;
        s += v;
        s2 += v * v;
    }
    atomicAdd(&colSum[c], s);
    atomicAdd(&colSumSq[c], s2);
}

// Tiny kernel: fold BN into per-column scale/shift. 1 block x 128 threads.
__global__ void bn_coef_kernel(const float* __restrict__ colSum,
                               const float* __restrict__ colSumSq,
                               const float* __restrict__ g,
                               const float* __restrict__ be,
                               float* __restrict__ scale,
                               float* __restrict__ shift,
                               int nRows) {
    const int c = threadIdx.x;
    const float inv = 1.0f / (float)nRows;
    const float mu = colSum[c] * inv;
    float var = colSumSq[c] * inv - mu * mu;
    if (var < 0.0f) var = 0.0f;                    // numerical guard
    const float rs = rsqrtf(var + EPS_BN);
    const float sc = rs * g[c];
    scale[c] = sc;
    shift[c] = fmaf(-mu, sc, be[c]);
}

// BN apply (+ optional ELU): out = y*scale[c] + shift[c]
template <bool DO_ELU>
__global__ void bn_apply_kernel(const float* __restrict__ Y,
                                const float* __restrict__ scale,
                                const float* __restrict__ shift,
                                float* __restrict__ out,
                                long long total) {
    long long i = (long long)blockIdx.x * blockDim.x + threadIdx.x;
    if (i >= total) return;
    const int c = (int)(i & (D_FEAT - 1));
    float v = fmaf(Y[i], scale[c], shift[c]);
    if (DO_ELU) v = v > 0.0f ? v : expm1f(v);      // jax.nn.elu, alpha=1
    out[i] = v;
}

// ---------------------------------------------------------------------------
// Orchestration
// ---------------------------------------------------------------------------
extern "C" void kernel_launch(void* const* d_in, const int* in_sizes, int n_in,
                              void* d_out, int out_size, void* d_ws, size_t ws_size,
                              hipStream_t stream) {
    // Inputs in setup_inputs() order
    const float* seq  = (const float*)d_in[0];
    const int*   esrc = (const int*)  d_in[1];
    const int*   edst = (const int*)  d_in[2];
    const float* evl  = (const float*)d_in[3];
    const float* W0   = (const float*)d_in[4];
    const float* b0   = (const float*)d_in[5];
    const float* a0   = (const float*)d_in[6];   // scalar (1 elem)
    const float* g0   = (const float*)d_in[7];
    const float* be0  = (const float*)d_in[8];
    const float* W1   = (const float*)d_in[9];
    const float* b1   = (const float*)d_in[10];
    const float* a1   = (const float*)d_in[11];
    const float* g1   = (const float*)d_in[12];
    const float* be1  = (const float*)d_in[13];

    const int N  = in_sizes[0] / D_FEAT;          // 50000
    const int E  = in_sizes[1];                   // 800000
    const long long ND = (long long)N * D_FEAT;

    // Workspace: h | agg | go0 | sum(128) | sumsq(128) | scale(128) | shift(128)
    // ~76.8 MB total; ws_size is assumed sufficient.
    float* h     = (float*)d_ws;
    float* agg   = h   + ND;
    float* go0   = agg + ND;
    float* sum   = go0 + ND;
    float* sumsq = sum   + D_FEAT;
    float* scale = sumsq + D_FEAT;
    float* shift = scale + D_FEAT;

    const int nTiles   = N / 16;
    const int remStart = nTiles * 16;
    const int remRows  = N - remStart;
    const int scatGrid = (E + 7) / 8;
    const int statGrid = (N + ROWS_PER_BLOCK - 1) / ROWS_PER_BLOCK;
    const int applyGrid = (int)((ND + 255) / 256);

    // ---------------- Layer 0 ----------------
    gemm_bias_wmma_kernel<<<nTiles, 256, 0, stream>>>(seq, W0, b0, h, nTiles);
    if (remRows > 0) {
        int t = remRows * D_FEAT;
        gemm_rem_kernel<<<(t + 255) / 256, 256, 0, stream>>>(seq, W0, b0, h, remStart, N);
    }
    hipMemsetAsync(agg, 0, (size_t)ND * sizeof(float), stream);
    hipMemsetAsync(sum, 0, 2 * D_FEAT * sizeof(float), stream);
    scatter_kernel<<<scatGrid, 256, 0, stream>>>(h, esrc, edst, evl, agg, E);
    prelu_res_stats_kernel<<<statGrid, D_FEAT, 0, stream>>>(agg, nullptr, a0, sum, sumsq, N);
    bn_coef_kernel<<<1, D_FEAT, 0, stream>>>(sum, sumsq, g0, be0, scale, shift, N);
    bn_apply_kernel<false><<<applyGrid, 256, 0, stream>>>(agg, scale, shift, go0, ND);

    // ---------------- Layer 1 (residual + ELU) ----------------
    gemm_bias_wmma_kernel<<<nTiles, 256, 0, stream>>>(go0, W1, b1, h, nTiles);
    if (remRows > 0) {
        int t = remRows * D_FEAT;
        gemm_rem_kernel<<<(t + 255) / 256, 256, 0, stream>>>(go0, W1, b1, h, remStart, N);
    }
    hipMemsetAsync(agg, 0, (size_t)ND * sizeof(float), stream);
    hipMemsetAsync(sum, 0, 2 * D_FEAT * sizeof(float), stream);
    scatter_kernel<<<scatGrid, 256, 0, stream>>>(h, esrc, edst, evl, agg, E);
    prelu_res_stats_kernel<<<statGrid, D_FEAT, 0, stream>>>(agg, go0, a1, sum, sumsq, N);
    bn_coef_kernel<<<1, D_FEAT, 0, stream>>>(sum, sumsq, g1, be1, scale, shift, N);
    bn_apply_kernel<true><<<applyGrid, 256, 0, stream>>>(agg, scale, shift, (float*)d_out, ND);
}